// Net_47356309406114
// MI455X (gfx1250) — compile-verified
//
#include <hip/hip_runtime.h>
#include <math.h>

typedef __attribute__((ext_vector_type(2))) float v2f;
typedef __attribute__((ext_vector_type(8))) float v8f;

#define N_NODES   50000
#define N_EDGES   800000
#define F_IN      300
#define HEADS     4
#define PER_HEAD  32
#define HIDDEN    128
#define N_CLASSES 9
#define NC_PAD    16
#define NEG_SLOPE 0.2f
#define EDGES_PER_BLOCK 4
#define KC        32          // K-chunk per LDS stage (multiple of 4)
#define MAX_FOUTP 128

// -------- float atomic max via sign-aware integer atomics (init to -inf) ----
__device__ __forceinline__ void atomicMaxF32(float* addr, float val) {
    if (val >= 0.0f) {
        atomicMax((int*)addr, __float_as_int(val));
    } else {
        atomicMin((unsigned int*)addr, (unsigned int)__float_as_int(val));
    }
}

// ---------------------------------------------------------------------------
__global__ void fill_kernel(float* __restrict__ p, float v, int n) {
    int i = blockIdx.x * blockDim.x + threadIdx.x;
    if (i < n) p[i] = v;
}

// Zero-padded weight copy: Wp[finp x foutp] <- W[fin x fout], rest = 0
__global__ void pad_weight_kernel(const float* __restrict__ W, float* __restrict__ Wp,
                                  int fin, int fout, int finp, int foutp) {
    int i = blockIdx.x * blockDim.x + threadIdx.x;
    if (i >= finp * foutp) return;
    int rr = i / foutp;
    int cc = i - rr * foutp;
    Wp[i] = (rr < fin && cc < fout) ? W[(size_t)rr * fout + cc] : 0.0f;
}

// ---------------------------------------------------------------------------
// C[n x foutp] = A[n x fin] * Bp[finp x foutp]   (Bp pre-padded with zeros)
// grid.x = n/16 row tiles; blockDim = (32, foutp/16): one wave per 16-col tile.
// V_WMMA_F32_16X16X4_F32 lane layouts:
//   A 16x4 : lane(0-15)=M, vgpr0/1 = K0,K1 ; lanes 16-31 = K2,K3
//   B 4x16 : vgpr0/1 = rows K0,K1 (lanes 0-15, N=lane) / K2,K3 (lanes 16-31)
//   C 16x16: vgpr v -> M = v + 8*(lane>=16), N = lane&15
__global__ void gemm_wmma_kernel(const float* __restrict__ A,
                                 const float* __restrict__ Bp,
                                 float* __restrict__ C,
                                 int fin, int finp, int foutp) {
    __shared__ float As[16][KC + 4];
    __shared__ float Bs[KC][MAX_FOUTP + 4];
    const int lane = threadIdx.x;
    const int ct   = threadIdx.y;
    const int r    = lane & 15;
    const int hi   = lane >> 4;
    const int row0 = blockIdx.x * 16;
    const int col  = ct * 16 + r;
    const int tid  = ct * 32 + lane;
    const int nthr = blockDim.y * 32;

    v8f acc = {};
    for (int kb = 0; kb < finp; kb += KC) {
        // A tile 16xKC (zero-pad K tail), coalesced
        for (int idx = tid; idx < 16 * KC; idx += nthr) {
            int rr = idx / KC;
            int cc = idx - rr * KC;
            int k  = kb + cc;
            As[rr][cc] = (k < fin) ? A[(size_t)(row0 + rr) * fin + k] : 0.0f;
        }
        // B chunk KC x foutp from padded weights, coalesced, no guards
        for (int idx = tid; idx < KC * foutp; idx += nthr) {
            int rr = idx / foutp;
            int cc = idx - rr * foutp;
            Bs[rr][cc] = Bp[(size_t)(kb + rr) * foutp + cc];
        }
        __syncthreads();
        #pragma unroll
        for (int k2 = 0; k2 < KC; k2 += 4) {
            v2f av, bv;
            av.x = As[r][k2 + 2 * hi];
            av.y = As[r][k2 + 2 * hi + 1];
            bv.x = Bs[k2 + 2 * hi][col];
            bv.y = Bs[k2 + 2 * hi + 1][col];
            acc = __builtin_amdgcn_wmma_f32_16x16x4_f32(
                false, av, false, bv, (short)0, acc, false, false);
        }
        __syncthreads();
    }
    #pragma unroll
    for (int v = 0; v < 8; ++v) {
        C[(size_t)(row0 + v + 8 * hi) * foutp + col] = acc[v];
    }
}

// ---------------------------------------------------------------------------
// per-(node,head) attention logits: asrc = <H[n,h,:], att_src[h,:]>, same adst
__global__ void att_kernel(const float* __restrict__ Hf,
                           const float* __restrict__ a_s,
                           const float* __restrict__ a_d,
                           float* __restrict__ asrc, float* __restrict__ adst,
                           int n, int heads, int chead, int ldh) {
    int i = blockIdx.x * blockDim.x + threadIdx.x;
    if (i >= n * heads) return;
    int node = i / heads;
    int h    = i - node * heads;
    const float* row = Hf + (size_t)node * ldh + h * chead;
    const float* w1  = a_s + h * chead;
    const float* w2  = a_d + h * chead;
    float s = 0.0f, d = 0.0f;
    for (int c = 0; c < chead; ++c) {
        float v = row[c];
        s += v * w1[c];
        d += v * w2[c];
    }
    asrc[i] = s;
    adst[i] = d;
}

// edge pass 1: e = leaky_relu(asrc[src]+adst[dst]); segment max into emax[dst]
__global__ void edge_pass1(const int* __restrict__ ei,
                           const float* __restrict__ asrc,
                           const float* __restrict__ adst,
                           float* __restrict__ estore, float* __restrict__ emax,
                           int nE, int heads) {
    int i = blockIdx.x * blockDim.x + threadIdx.x;
    if (i >= nE * heads) return;
    int e = i / heads;
    int h = i - e * heads;
    int src = ei[e];
    int dst = ei[nE + e];
    float a = asrc[src * heads + h] + adst[dst * heads + h];
    float v = (a > 0.0f) ? a : NEG_SLOPE * a;
    estore[i] = v;
    atomicMaxF32(&emax[dst * heads + h], v);
}

// edge pass 2: e_exp = exp(e - emax[dst]); segment sum into denom[dst]
__global__ void edge_pass2(const int* __restrict__ ei,
                           float* __restrict__ estore,
                           const float* __restrict__ emax,
                           float* __restrict__ denom, int nE, int heads) {
    int i = blockIdx.x * blockDim.x + threadIdx.x;
    if (i >= nE * heads) return;
    int e = i / heads;
    int h = i - e * heads;
    int dst = ei[nE + e];
    float ex = expf(estore[i] - emax[dst * heads + h]);
    estore[i] = ex;
    atomicAdd(&denom[dst * heads + h], ex);
}

// edge pass 3: out[dst,:] += (e_exp/denom[dst]) * H[src,:]  (coalesced 128-wide)
__global__ void edge_aggregate(const int* __restrict__ ei,
                               const float* __restrict__ estore,
                               const float* __restrict__ denom,
                               const float* __restrict__ Hf,
                               float* __restrict__ outp,
                               int nE, int heads, int chead, int ld) {
    int t = threadIdx.x;
    bool active = (t < heads * chead);
    int h = active ? (t / chead) : 0;
    int e0 = blockIdx.x * EDGES_PER_BLOCK;
    for (int j = 0; j < EDGES_PER_BLOCK; ++j) {
        int e = e0 + j;
        if (e >= nE) break;
        int src = ei[e];
        int dst = ei[nE + e];
        if (active) {
            float alpha = estore[e * heads + h] / denom[dst * heads + h];
            atomicAdd(&outp[(size_t)dst * ld + t], alpha * Hf[(size_t)src * ld + t]);
        }
    }
}

// out = elu(out + bias), in place, row width f
__global__ void bias_elu_kernel(float* __restrict__ p, const float* __restrict__ b,
                                int n, int f) {
    int i = blockIdx.x * blockDim.x + threadIdx.x;
    if (i >= n * f) return;
    float v = p[i] + b[i % f];
    p[i] = (v > 0.0f) ? v : expm1f(v);
}

// final: elu(agg + b3) then log_softmax over 9 classes -> d_out [N,9]
__global__ void final_kernel(const float* __restrict__ agg,
                             const float* __restrict__ b,
                             float* __restrict__ out, int n) {
    int i = blockIdx.x * blockDim.x + threadIdx.x;
    if (i >= n) return;
    float v[N_CLASSES];
    float m = -1e30f;
    #pragma unroll
    for (int c = 0; c < N_CLASSES; ++c) {
        float x = agg[(size_t)i * NC_PAD + c] + b[c];
        x = (x > 0.0f) ? x : expm1f(x);
        v[c] = x;
        m = fmaxf(m, x);
    }
    float s = 0.0f;
    #pragma unroll
    for (int c = 0; c < N_CLASSES; ++c) s += expf(v[c] - m);
    float lse = m + logf(s);
    #pragma unroll
    for (int c = 0; c < N_CLASSES; ++c)
        out[(size_t)i * N_CLASSES + c] = v[c] - lse;
}

// ---------------------------------------------------------------------------
static inline int cdiv(int a, int b) { return (a + b - 1) / b; }

static void run_gat_layer(const float* X, int fin,
                          const float* W, const float* a_s, const float* a_d,
                          const float* bias, int heads, int chead,
                          int fout, int foutp,
                          const int* ei,
                          float* Wpad, float* Hbuf, float* outbuf,
                          float* asrc, float* adst, float* emax, float* denom,
                          float* estore, bool do_bias_elu, hipStream_t stream) {
    int finp = cdiv(fin, KC) * KC;
    // 0) zero-padded weights -> no guards in GEMM inner loop
    pad_weight_kernel<<<cdiv(finp * foutp, 256), 256, 0, stream>>>(
        W, Wpad, fin, fout, finp, foutp);
    // 1) GEMM (WMMA, LDS-staged A and B tiles)
    dim3 gblk(32, foutp / 16);
    gemm_wmma_kernel<<<N_NODES / 16, gblk, 0, stream>>>(X, Wpad, Hbuf, fin, finp, foutp);
    // 2) attention logits
    att_kernel<<<cdiv(N_NODES * heads, 256), 256, 0, stream>>>(
        Hbuf, a_s, a_d, asrc, adst, N_NODES, heads, chead, foutp);
    // 3) init accumulators
    fill_kernel<<<cdiv(N_NODES * foutp, 256), 256, 0, stream>>>(outbuf, 0.0f, N_NODES * foutp);
    fill_kernel<<<cdiv(N_NODES * heads, 256), 256, 0, stream>>>(emax, -INFINITY, N_NODES * heads);
    fill_kernel<<<cdiv(N_NODES * heads, 256), 256, 0, stream>>>(denom, 0.0f, N_NODES * heads);
    // 4) edge passes
    edge_pass1<<<cdiv(N_EDGES * heads, 256), 256, 0, stream>>>(
        ei, asrc, adst, estore, emax, N_EDGES, heads);
    edge_pass2<<<cdiv(N_EDGES * heads, 256), 256, 0, stream>>>(
        ei, estore, emax, denom, N_EDGES, heads);
    int ablk = (foutp < 32) ? 32 : foutp;
    edge_aggregate<<<cdiv(N_EDGES, EDGES_PER_BLOCK), ablk, 0, stream>>>(
        ei, estore, denom, Hbuf, outbuf, N_EDGES, heads, chead, foutp);
    // 5) epilogue
    if (do_bias_elu) {
        bias_elu_kernel<<<cdiv(N_NODES * fout, 256), 256, 0, stream>>>(
            outbuf, bias, N_NODES, fout);
    }
}

extern "C" void kernel_launch(void* const* d_in, const int* in_sizes, int n_in,
                              void* d_out, int out_size, void* d_ws, size_t ws_size,
                              hipStream_t stream) {
    const float* x   = (const float*)d_in[0];
    const int*   ei  = (const int*)d_in[1];
    const float* W1  = (const float*)d_in[2];
    const float* as1 = (const float*)d_in[3];
    const float* ad1 = (const float*)d_in[4];
    const float* b1  = (const float*)d_in[5];
    const float* W2  = (const float*)d_in[6];
    const float* as2 = (const float*)d_in[7];
    const float* ad2 = (const float*)d_in[8];
    const float* b2  = (const float*)d_in[9];
    const float* W3  = (const float*)d_in[10];
    const float* as3 = (const float*)d_in[11];
    const float* ad3 = (const float*)d_in[12];
    const float* b3  = (const float*)d_in[13];
    float* out = (float*)d_out;

    // workspace layout (floats)
    float* ws     = (float*)d_ws;
    float* Q      = ws;                                   // [N,128]  GEMM output
    float* P      = Q      + (size_t)N_NODES * HIDDEN;    // [N,128]  agg / next input
    float* asrc   = P      + (size_t)N_NODES * HIDDEN;    // [N,4]
    float* adst   = asrc   + (size_t)N_NODES * HEADS;     // [N,4]
    float* emax   = adst   + (size_t)N_NODES * HEADS;     // [N,4]
    float* denom  = emax   + (size_t)N_NODES * HEADS;     // [N,4]
    float* estore = denom  + (size_t)N_NODES * HEADS;     // [E,4]
    float* Wpad   = estore + (size_t)N_EDGES * HEADS;     // [320,128] max padded W

    // Layer 1: x[N,300] -> P[N,128]
    run_gat_layer(x, F_IN, W1, as1, ad1, b1, HEADS, PER_HEAD, HIDDEN, HIDDEN,
                  ei, Wpad, Q, P, asrc, adst, emax, denom, estore, true, stream);
    // Layer 2: P[N,128] -> P[N,128]  (Q reused as GEMM buffer)
    run_gat_layer(P, HIDDEN, W2, as2, ad2, b2, HEADS, PER_HEAD, HIDDEN, HIDDEN,
                  ei, Wpad, Q, P, asrc, adst, emax, denom, estore, true, stream);
    // Layer 3: P[N,128] -> agg in P[N,16] (padded 9->16), bias+elu+log_softmax fused
    run_gat_layer(P, HIDDEN, W3, as3, ad3, b3, 1, N_CLASSES, N_CLASSES, NC_PAD,
                  ei, Wpad, Q, P, asrc, adst, emax, denom, estore, false, stream);
    final_kernel<<<cdiv(N_NODES, 256), 256, 0, stream>>>(P, b3, out, N_NODES);
}